// MoeMatmul_627065225256
// MI455X (gfx1250) — compile-verified
//
#include <hip/hip_runtime.h>
#include <stdint.h>

typedef float v2f __attribute__((ext_vector_type(2)));
typedef float v8f __attribute__((ext_vector_type(8)));
typedef unsigned int u32x4 __attribute__((ext_vector_type(4)));
typedef int i32x4 __attribute__((ext_vector_type(4)));
typedef int i32x8 __attribute__((ext_vector_type(8)));

constexpr int T = 8192;
constexpr int D = 2048;
constexpr int F = 8192;
constexpr int E = 8;

constexpr int BM = 128;
constexpr int BN = 128;
constexpr int BK = 32;
constexpr int MAX_MTILES = T / BM + E;   // 72 worst case after per-expert padding
constexpr int PERM_CAP  = T + E * BM;    // 9216

// LDS staging (floats): A rows stride 36 (32 data + 4 pad), B rows stride 136 (128 + 8 pad)
constexpr int A_STRIDE = 36;
constexpr int B_STRIDE = 136;
constexpr int A_STAGE  = BM * A_STRIDE;          // 4608 floats (18432 B)
constexpr int B_STAGE  = BK * B_STRIDE;          // 4352 floats (17408 B)
constexpr int STAGE    = A_STAGE + B_STAGE;      // 8960 floats (35840 B)

// workspace (int) layout
constexpr int WS_COUNTS   = 0;    // [8]
constexpr int WS_CURSORS  = 8;    // [8]
constexpr int WS_POFF     = 16;   // [9] padded offsets
constexpr int WS_NTILES   = 25;   // [1]
constexpr int WS_TILE_EXP = 32;   // [MAX_MTILES]
constexpr int WS_PERM     = 128;  // [PERM_CAP]

__global__ void moe_init(int* __restrict__ ws) {
  int i = blockIdx.x * blockDim.x + threadIdx.x;
  if (i < 32) ws[i] = 0;                   // counts + cursors
  if (i < PERM_CAP) ws[WS_PERM + i] = -1;  // padded slots -> "no token"
}

__global__ void moe_hist(const int* __restrict__ eid, int* __restrict__ ws) {
  int t = blockIdx.x * blockDim.x + threadIdx.x;
  if (t < T) atomicAdd(&ws[WS_COUNTS + eid[t]], 1);
}

__global__ void moe_prefix(int* __restrict__ ws) {
  if (threadIdx.x == 0 && blockIdx.x == 0) {
    int off = 0;
    for (int e = 0; e < E; ++e) {
      ws[WS_POFF + e] = off;
      int tiles = (ws[WS_COUNTS + e] + BM - 1) / BM;
      for (int j = 0; j < tiles; ++j) ws[WS_TILE_EXP + off / BM + j] = e;
      off += tiles * BM;
    }
    ws[WS_POFF + E] = off;
    ws[WS_NTILES]   = off / BM;
  }
}

__global__ void moe_scatter(const int* __restrict__ eid, int* __restrict__ ws) {
  int t = blockIdx.x * blockDim.x + threadIdx.x;
  if (t < T) {
    int e   = eid[t];
    int pos = ws[WS_POFF + e] + atomicAdd(&ws[WS_CURSORS + e], 1);
    ws[WS_PERM + pos] = t;
  }
}

// Compiler fence: escape the LDS pointer + memory clobber so the optimizer
// treats TDM (tensor_load_to_lds) as having written the staging buffer.
__device__ __forceinline__ void lds_written_fence(float* p) {
  asm volatile("" : "+v"(p) : : "memory");
}

// Grouped GEMM with TDM double-buffered staging.
// 256 threads = 8 wave32; wave (wm,wn) owns a 32x64 sub-tile = 2x4 WMMA tiles.
__launch_bounds__(256)
__global__ void moe_gemm(const float* __restrict__ state,
                         const float* __restrict__ w,
                         const int*   __restrict__ ws,
                         float*       __restrict__ out) {
  int mtile = blockIdx.y;
  if (mtile >= ws[WS_NTILES]) return;   // uniform exit
  int expert = ws[WS_TILE_EXP + mtile];
  const float* wb = w + (size_t)expert * D * F;
  int n0 = blockIdx.x * BN;

  __shared__ float lds[2 * STAGE];      // sole shared block -> LDS byte offset 0

  const int tid   = threadIdx.x;
  const int lane  = tid & 31;
  const int wave  = tid >> 5;          // wave32
  const int wm    = (wave & 3) * 32;   // 0,32,64,96
  const int wn    = (wave >> 2) * 64;  // 0,64
  const int lmod  = lane & 15;
  const int lhalf = lane >> 4;         // 0: k,k+1 / M 0-7 ; 1: k+2,k+3 / M 8-15

  const int* perm = ws + WS_PERM + mtile * BM;

  // --- TDM descriptor constants ---------------------------------------------
  // Group1 for A (gather): data_size=4B(code2), pad_enable, pad_interval=32dw(code4),
  // pad_amount=4dw(code3); tensor_dim0=D, tensor_dim1=T, tile_dim0=32, tile_dim1=16 idx,
  // stride0=D.
  const i32x8 g1a = { (int)((2u << 16) | (1u << 20) | (4u << 22) | (3u << 25)),
                      (int)((unsigned)(D & 0xFFFF) << 16),
                      (int)((unsigned)(D >> 16) | ((unsigned)(T & 0xFFFF) << 16)),
                      (int)((unsigned)(T >> 16) | (32u << 16)),
                      16,            // tile_dim1 = 16 valid indices, tile_dim2 = 0
                      D,             // tensor_dim0_stride
                      0, 0 };
  // Group1 for B (dense 2D): pad_interval=128dw(code6), pad_amount=8dw(code7);
  // tensor_dim0=F, tensor_dim1=D, tile_dim0=BN, tile_dim1=BK, stride0=F.
  const i32x8 g1b = { (int)((2u << 16) | (1u << 20) | (6u << 22) | (7u << 25)),
                      (int)((unsigned)(F & 0xFFFF) << 16),
                      (int)((unsigned)(F >> 16) | ((unsigned)(D & 0xFFFF) << 16)),
                      (int)((unsigned)(D >> 16) | ((unsigned)BN << 16)),
                      BK,
                      F,
                      0, 0 };
  const i32x4 zero4 = { 0, 0, 0, 0 };
  const i32x8 zero8 = { 0, 0, 0, 0, 0, 0, 0, 0 };

  // Gather indices (8 descriptors x 16 rows), built once in lane 0.
  i32x4 g2v[8], g3v[8];
  if (tid == 0) {
#pragma unroll
    for (int d = 0; d < 8; ++d) {
      int idx[16];
#pragma unroll
      for (int i = 0; i < 16; ++i) {
        int tk = perm[d * 16 + i];
        idx[i] = tk < 0 ? 0 : tk;   // padded rows gather row 0; output masked later
      }
      g2v[d] = (i32x4){ idx[0] | (idx[1] << 16),  idx[2] | (idx[3] << 16),
                        idx[4] | (idx[5] << 16),  idx[6] | (idx[7] << 16) };
      g3v[d] = (i32x4){ idx[8] | (idx[9] << 16),  idx[10] | (idx[11] << 16),
                        idx[12] | (idx[13] << 16), idx[14] | (idx[15] << 16) };
    }
  }

  auto issue_stage = [&](int buf, int k0) {
    const unsigned stage_base = (unsigned)(buf * STAGE * 4);
    // A: 8 gather descriptors, 16 rows x 32 floats each, LDS stride 36 floats
    uint64_t ga = (uint64_t)(uintptr_t)state + (uint64_t)k0 * 4u;
    const unsigned aw2 = (unsigned)ga;
    const unsigned aw3 = (unsigned)((ga >> 32) & 0x01FFFFFFu) | (2u << 30);
#pragma unroll
    for (int d = 0; d < 8; ++d) {
      u32x4 g0 = { 1u | (1u << 31),                                   // count=1, gather, 16-bit idx
                   stage_base + (unsigned)(d * 16 * A_STRIDE * 4),    // lds_addr
                   aw2, aw3 };
      __builtin_amdgcn_tensor_load_to_lds(g0, g1a, g2v[d], g3v[d], zero8, 0);
    }
    // B: one 2D descriptor, 32 rows x 128 floats, LDS stride 136 floats
    uint64_t gb = (uint64_t)(uintptr_t)wb + ((uint64_t)k0 * F + (unsigned)n0) * 4u;
    u32x4 g0b = { 1u,
                  stage_base + (unsigned)(A_STAGE * 4),
                  (unsigned)gb,
                  (unsigned)((gb >> 32) & 0x01FFFFFFu) | (2u << 30) };
    __builtin_amdgcn_tensor_load_to_lds(g0b, g1b, zero4, zero4, zero8, 0);
  };

  v8f acc[2][4];
#pragma unroll
  for (int i = 0; i < 2; ++i)
#pragma unroll
    for (int j = 0; j < 4; ++j)
#pragma unroll
      for (int r = 0; r < 8; ++r) acc[i][j][r] = 0.0f;

  if (tid == 0) issue_stage(0, 0);     // prime the pipeline

  for (int k0 = 0, c = 0; k0 < D; k0 += BK, ++c) {
    if (tid == 0) {
      if (k0 + BK < D) {
        issue_stage((c + 1) & 1, k0 + BK);          // kick next stage (9 TDM ops)
        __builtin_amdgcn_s_wait_tensorcnt(9);       // in-order: stage c resident
      } else {
        __builtin_amdgcn_s_wait_tensorcnt(0);
      }
    }
    __syncthreads();
    lds_written_fence(lds);            // TDM wrote LDS: force fragment reloads

    const float* As = lds + (c & 1) * STAGE;
    const float* Bs = As + A_STAGE;

#pragma unroll
    for (int kk = 0; kk < BK; kk += 4) {
      const int kb = kk + lhalf * 2;
      v2f a0, a1;
      a0.x = As[(wm + lmod) * A_STRIDE + kb];
      a0.y = As[(wm + lmod) * A_STRIDE + kb + 1];
      a1.x = As[(wm + 16 + lmod) * A_STRIDE + kb];
      a1.y = As[(wm + 16 + lmod) * A_STRIDE + kb + 1];
#pragma unroll
      for (int j = 0; j < 4; ++j) {
        const int col = wn + j * 16 + lmod;
        v2f b;
        b.x = Bs[kb * B_STRIDE + col];
        b.y = Bs[(kb + 1) * B_STRIDE + col];
        acc[0][j] = __builtin_amdgcn_wmma_f32_16x16x4_f32(
            false, a0, false, b, (short)0, acc[0][j], false, false);
        acc[1][j] = __builtin_amdgcn_wmma_f32_16x16x4_f32(
            false, a1, false, b, (short)0, acc[1][j], false, false);
      }
    }
    __syncthreads();
  }

  // scatter results back to token rows (skip padded slots)
#pragma unroll
  for (int i = 0; i < 2; ++i) {
#pragma unroll
    for (int r = 0; r < 8; ++r) {
      int mloc = wm + i * 16 + lhalf * 8 + r;   // C layout: vgpr r, lanes16-31 -> M=r+8
      int tok  = perm[mloc];
      if (tok >= 0) {
        float* orow = out + (size_t)tok * F + n0 + wn;
#pragma unroll
        for (int j = 0; j < 4; ++j)
          orow[j * 16 + lmod] = acc[i][j][r];
      }
    }
  }
}

extern "C" void kernel_launch(void* const* d_in, const int* in_sizes, int n_in,
                              void* d_out, int out_size, void* d_ws, size_t ws_size,
                              hipStream_t stream) {
  const float* state = (const float*)d_in[0];
  const int*   eids  = (const int*)d_in[1];
  const float* w     = (const float*)d_in[2];
  float*       out   = (float*)d_out;
  int*         ws    = (int*)d_ws;

  moe_init<<<dim3((PERM_CAP + 255) / 256), 256, 0, stream>>>(ws);
  moe_hist<<<dim3(T / 256), 256, 0, stream>>>(eids, ws);
  moe_prefix<<<dim3(1), 32, 0, stream>>>(ws);
  moe_scatter<<<dim3(T / 256), 256, 0, stream>>>(eids, ws);
  moe_gemm<<<dim3(F / BN, MAX_MTILES), 256, 0, stream>>>(state, w, ws, out);
}